// GAT_86019605004975
// MI455X (gfx1250) — compile-verified
//
#include <hip/hip_runtime.h>
#include <hip/hip_bf16.h>
#include <stdint.h>

// Only check AMDGCN builtins during the device pass; the host (x86) pass
// reports 0 for __has_builtin of target builtins.
#if defined(__HIP_DEVICE_COMPILE__)
#if !__has_builtin(__builtin_amdgcn_wmma_f32_16x16x4_f32)
#error "device pass: missing __builtin_amdgcn_wmma_f32_16x16x4_f32"
#endif
#endif

typedef float v2f __attribute__((ext_vector_type(2)));
typedef float v8f __attribute__((ext_vector_type(8)));
typedef unsigned int v4u __attribute__((ext_vector_type(4)));
typedef int v8i __attribute__((ext_vector_type(8)));
typedef int v4i __attribute__((ext_vector_type(4)));

#define NEG_SLOPE 0.2f

// ---------- helpers ----------
__device__ __forceinline__ unsigned fkey(float f) {
  unsigned u = __float_as_uint(f);
  return (u & 0x80000000u) ? ~u : (u | 0x80000000u);   // monotonic order-preserving map
}
__device__ __forceinline__ float fdec(unsigned k) {
  return __uint_as_float((k & 0x80000000u) ? (k & 0x7fffffffu) : ~k);
}
// edges 0..E-1 come from edge_index, edges E..E+N-1 are self loops
__device__ __forceinline__ void edge_nodes(const int* __restrict__ ei, int E, long long e,
                                           int& s, int& d) {
  if (e < (long long)E) { s = ei[e]; d = ei[(long long)E + e]; }
  else                  { s = d = (int)(e - E); }
}

// ---------- TDM: DMA a 2D tile [tile_d1 rows x tile_d0 cols] of 4-byte data
// from a row-major tensor [tensor_d1 x tensor_d0] (row stride = stride0 elems)
// into LDS at byte offset lds_addr. D# layout per CDNA5 ISA 8.3/8.4.
// This toolchain exposes the 6-arg builtin:
//   (uint32x4 g0, int32x8 g1, int32x4 g2, int32x4 g3, int32x8 pad, i32 cpol)
__device__ __forceinline__ void tdm_load_2d_to_lds(unsigned lds_addr, const void* gaddr,
                                                   unsigned tensor_d0, unsigned tensor_d1,
                                                   unsigned tile_d0, unsigned tile_d1,
                                                   unsigned stride0) {
#if __has_builtin(__builtin_amdgcn_tensor_load_to_lds) && __has_builtin(__builtin_amdgcn_s_wait_tensorcnt)
  const unsigned long long ga = (unsigned long long)gaddr;
  v4u g0;
  g0[0] = 1u;                                             // count=1, user descriptor
  g0[1] = lds_addr;                                       // lds_addr (bytes)
  g0[2] = (unsigned)(ga & 0xffffffffu);                   // global_addr[31:0]
  g0[3] = (unsigned)((ga >> 32) & 0x01ffffffu) | (2u << 30);  // addr[56:32] | type=2
  v8i g1;
  g1[0] = (int)(2u << 16);                                // data_size=4B; no mask/flags
  g1[1] = (int)((tensor_d0 & 0xffffu) << 16);             // abar_addr=0 | dim0[15:0]
  g1[2] = (int)(((tensor_d0 >> 16) & 0xffffu) | ((tensor_d1 & 0xffffu) << 16));
  g1[3] = (int)(((tensor_d1 >> 16) & 0xffffu) | ((tile_d0 & 0xffffu) << 16));
  g1[4] = (int)(tile_d1 & 0xffffu);                       // tile_dim1 | tile_dim2=0
  g1[5] = (int)stride0;                                   // dim0_stride[31:0]
  g1[6] = 0;                                              // dim0_stride hi | dim1_stride lo
  g1[7] = 0;
  v4i z4; z4[0] = 0; z4[1] = 0; z4[2] = 0; z4[3] = 0;     // 2D: groups 2/3 unused
  v8i z8;
#pragma unroll
  for (int i = 0; i < 8; ++i) z8[i] = 0;
  __builtin_amdgcn_tensor_load_to_lds(g0, g1, z4, z4, z8, 0);
  __builtin_amdgcn_s_wait_tensorcnt(0);
#else
  (void)lds_addr; (void)gaddr; (void)tensor_d0; (void)tensor_d1;
  (void)tile_d0; (void)tile_d1; (void)stride0;
#endif
}

// ---------- fp32 WMMA GEMM: C[M,N] = A[M,K] * B[K,N] ----------
// block = 256 threads = 8 waves; each wave -> 16 rows x 64 cols (4 accumulators of
// v_wmma_f32_16x16x4_f32). B panel [K x 64] staged in LDS once per block via TDM.
// grid = (ceil(M/128), N/64), dynamic LDS = K*64*4 bytes. N%64==0, K%4==0.
__global__ __launch_bounds__(256)
void gemm_f32_wmma(const float* __restrict__ A, const float* __restrict__ B,
                   float* __restrict__ C, int M, int N, int K) {
  extern __shared__ float Bl[];                   // [K][64]
  const int lane = threadIdx.x & 31;
  const int wave = threadIdx.x >> 5;
  const int half = lane >> 4;                     // 0: lanes 0-15, 1: lanes 16-31
  const int l15  = lane & 15;
  const int rowTile = blockIdx.x * 128 + wave * 16;
  const int colTile = blockIdx.y * 64;

  // Stage B[0:K, colTile:colTile+64] -> LDS.
#if defined(__HIP_DEVICE_COMPILE__) && __has_builtin(__builtin_amdgcn_tensor_load_to_lds)
  if (threadIdx.x < 32) {                         // one wave issues the DMA + waits
    tdm_load_2d_to_lds((unsigned)(size_t)(void*)Bl, (const void*)(B + colTile),
                       (unsigned)N, (unsigned)K, 64u, (unsigned)K, (unsigned)N);
  }
#else
  for (int i = threadIdx.x; i < K * 64; i += 256)
    Bl[i] = B[(size_t)(i >> 6) * N + colTile + (i & 63)];
#endif
  __syncthreads();

  int aRow = rowTile + l15;
  if (aRow >= M) aRow = M - 1;                    // clamp tail (stores are guarded)
  const float* __restrict__ Arow = A + (size_t)aRow * K;

  v8f acc[4];
#pragma unroll
  for (int j = 0; j < 4; ++j)
#pragma unroll
    for (int r = 0; r < 8; ++r) acc[j][r] = 0.0f;

  for (int k0 = 0; k0 < K; k0 += 4) {
    const int k = k0 + half * 2;                  // A 16x4 f32 layout: half-wave holds K pair
    v2f a;
    a.x = Arow[k];
    a.y = Arow[k + 1];
#pragma unroll
    for (int j = 0; j < 4; ++j) {
      const int n = j * 16 + l15;                 // col within the 64-wide LDS panel
      v2f b;
      b.x = Bl[k * 64 + n];
      b.y = Bl[(k + 1) * 64 + n];
      acc[j] = __builtin_amdgcn_wmma_f32_16x16x4_f32(
          /*neg_a=*/false, a, /*neg_b=*/false, b,
          /*c_mod=*/(short)0, acc[j], /*reuse_a=*/false, /*reuse_b=*/false);
    }
  }
  // D layout: vgpr r -> row rowTile + r + 8*half, col = colTile + j*16 + l15
#pragma unroll
  for (int j = 0; j < 4; ++j) {
    const int n = colTile + j * 16 + l15;
#pragma unroll
    for (int r = 0; r < 8; ++r) {
      const int row = rowTile + r + 8 * half;
      if (row < M) C[(size_t)row * N + n] = acc[j][r];
    }
  }
}

// ---------- per-(node,head) attention score dots: one wave per (n,h) ----------
__global__ __launch_bounds__(256)
void node_scores(const float* __restrict__ h, const float* __restrict__ a_src,
                 const float* __restrict__ a_dst, float* __restrict__ s_src,
                 float* __restrict__ s_dst, int Nn, int H, int C) {
  const int wid  = (int)((blockIdx.x * (unsigned)blockDim.x + threadIdx.x) >> 5);
  const int lane = threadIdx.x & 31;
  if (wid >= Nn * H) return;
  const int n = wid / H, hh = wid % H;
  const float* __restrict__ hp = h + ((size_t)n * H + hh) * C;
  const float* __restrict__ as = a_src + (size_t)hh * C;
  const float* __restrict__ ad = a_dst + (size_t)hh * C;
  float ss = 0.f, sd = 0.f;
  for (int c = lane; c < C; c += 32) {
    const float v = hp[c];
    ss += v * as[c];
    sd += v * ad[c];
  }
#pragma unroll
  for (int off = 16; off > 0; off >>= 1) {
    ss += __shfl_down(ss, off, 32);
    sd += __shfl_down(sd, off, 32);
  }
  if (lane == 0) { s_src[wid] = ss; s_dst[wid] = sd; }
}

// ---------- segment max of leaky_relu(edge score) over dst ----------
__global__ __launch_bounds__(256)
void edge_segmax(const int* __restrict__ ei, int E, int Nn,
                 const float* __restrict__ s_src, const float* __restrict__ s_dst,
                 unsigned* __restrict__ emax, int H) {
  const long long idx = blockIdx.x * (long long)blockDim.x + threadIdx.x;
  const long long tot = (long long)(E + Nn) * H;
  if (idx >= tot) return;
  const int hh = (int)(idx % H);
  const long long e = idx / H;
  int s, d; edge_nodes(ei, E, e, s, d);
  float sc = s_src[(size_t)s * H + hh] + s_dst[(size_t)d * H + hh];
  sc = sc > 0.f ? sc : NEG_SLOPE * sc;
  atomicMax(&emax[(size_t)d * H + hh], fkey(sc));
}

// ---------- w = exp(e - max); denom = segment_sum(w) ----------
__global__ __launch_bounds__(256)
void edge_expsum(const int* __restrict__ ei, int E, int Nn,
                 const float* __restrict__ s_src, const float* __restrict__ s_dst,
                 const unsigned* __restrict__ emax, float* __restrict__ wbuf,
                 float* __restrict__ denom, int H) {
  const long long idx = blockIdx.x * (long long)blockDim.x + threadIdx.x;
  const long long tot = (long long)(E + Nn) * H;
  if (idx >= tot) return;
  const int hh = (int)(idx % H);
  const long long e = idx / H;
  int s, d; edge_nodes(ei, E, e, s, d);
  float sc = s_src[(size_t)s * H + hh] + s_dst[(size_t)d * H + hh];
  sc = sc > 0.f ? sc : NEG_SLOPE * sc;
  const float m = fdec(emax[(size_t)d * H + hh]);
  const float w = __expf(sc - m);
  wbuf[idx] = w;                                   // layout [e, h]
  atomicAdd(&denom[(size_t)d * H + hh], w);
}

// ---------- alpha = w / (denom[dst] + eps), in place ----------
__global__ __launch_bounds__(256)
void edge_norm(const int* __restrict__ ei, int E, int Nn, float* __restrict__ wbuf,
               const float* __restrict__ denom, int H) {
  const long long idx = blockIdx.x * (long long)blockDim.x + threadIdx.x;
  const long long tot = (long long)(E + Nn) * H;
  if (idx >= tot) return;
  const int hh = (int)(idx % H);
  const long long e = idx / H;
  int s, d; edge_nodes(ei, E, e, s, d);
  wbuf[idx] = wbuf[idx] / (denom[(size_t)d * H + hh] + 1e-16f);
}

// ---------- out[dst,h,:] += alpha * h[src,h,:] ----------
__global__ __launch_bounds__(256)
void aggregate(const int* __restrict__ ei, int E, int Nn, const float* __restrict__ h,
               const float* __restrict__ alpha, float* __restrict__ out, int H, int C) {
  const long long idx = blockIdx.x * (long long)blockDim.x + threadIdx.x;
  const long long tot = (long long)(E + Nn) * H * C;
  if (idx >= tot) return;
  const int c = (int)(idx % C);
  const long long t = idx / C;
  const int hh = (int)(t % H);
  const long long e = t / H;
  int s, d; edge_nodes(ei, E, e, s, d);
  const float a = alpha[e * H + hh];
  atomicAdd(&out[((size_t)d * H + hh) * C + c],
            h[((size_t)s * H + hh) * C + c] * a);
}

// ---------- elementwise ----------
__global__ __launch_bounds__(256)
void bias_elu_inplace(float* __restrict__ x, const float* __restrict__ b, size_t n, int F) {
  const size_t i = (size_t)blockIdx.x * blockDim.x + threadIdx.x;
  if (i >= n) return;
  const float v = x[i] + b[i % F];
  x[i] = v > 0.f ? v : (__expf(v) - 1.f);
}

__global__ __launch_bounds__(256)
void residual_out(const float* __restrict__ x, const float* __restrict__ o1,
                  const float* __restrict__ b, float* __restrict__ out, size_t n, int F) {
  const size_t i = (size_t)blockIdx.x * blockDim.x + threadIdx.x;
  if (i >= n) return;
  out[i] = x[i] + o1[i] + b[i % F];
}

__global__ __launch_bounds__(256)
void zero_f32(float* __restrict__ p, size_t n) {
  const size_t i = (size_t)blockIdx.x * blockDim.x + threadIdx.x;
  if (i < n) p[i] = 0.f;
}

// ---------- launch ----------
static inline unsigned blocks_for(long long n, int bs) {
  return (unsigned)((n + bs - 1) / bs);
}

extern "C" void kernel_launch(void* const* d_in, const int* in_sizes, int n_in,
                              void* d_out, int out_size, void* d_ws, size_t ws_size,
                              hipStream_t stream) {
  const float* x     = (const float*)d_in[0];
  const float* W0    = (const float*)d_in[1];
  const float* asrc0 = (const float*)d_in[2];
  const float* adst0 = (const float*)d_in[3];
  const float* b0    = (const float*)d_in[4];
  const float* W1    = (const float*)d_in[5];
  const float* asrc1 = (const float*)d_in[6];
  const float* adst1 = (const float*)d_in[7];
  const float* b1    = (const float*)d_in[8];
  const int*   ei    = (const int*)d_in[9];

  const int IN = 128, HID = 128, OUTF = 128, H = 2;   // matches reference shapes
  const int Nn = in_sizes[0] / IN;                    // 50000
  const int E  = in_sizes[9] / 2;                     // 800000
  const long long ET = (long long)E + Nn;             // + self loops

  // workspace layout (floats). Total ~= 2*N*256 + N*8 + 2*ET + N*4 + ET  (~115 MB)
  float* ws = (float*)d_ws;
  size_t off = 0;
  float*    H0buf = ws + off; off += (size_t)Nn * H * HID;   // h0 [N,2,128]; reused by layer 1
  float*    OUT0  = ws + off; off += (size_t)Nn * H * HID;   // out0 -> h1 input (in-place elu)
  float*    SS0   = ws + off; off += (size_t)Nn * H;
  float*    SD0   = ws + off; off += (size_t)Nn * H;
  unsigned* EM0   = (unsigned*)(ws + off); off += (size_t)Nn * H;
  float*    DN0   = ws + off; off += (size_t)Nn * H;
  float*    W0b   = ws + off; off += (size_t)ET * H;
  float*    SS1   = ws + off; off += (size_t)Nn;
  float*    SD1   = ws + off; off += (size_t)Nn;
  unsigned* EM1   = (unsigned*)(ws + off); off += (size_t)Nn;
  float*    DN1   = ws + off; off += (size_t)Nn;
  float*    W1b   = ws + off; off += (size_t)ET;
  // layer-1 matrices alias the (then-dead) H0 region:
  float* H1buf = H0buf;                    // h1pre [N,128]
  float* OUT1  = H0buf + (size_t)Nn * OUTF;

  const dim3 blk(256);

  // ===== layer 0 =====
  zero_f32<<<blocks_for((long long)Nn * H * HID, 256), blk, 0, stream>>>(OUT0, (size_t)Nn * H * HID);
  zero_f32<<<blocks_for((long long)Nn * H, 256), blk, 0, stream>>>((float*)EM0, (size_t)Nn * H);
  zero_f32<<<blocks_for((long long)Nn * H, 256), blk, 0, stream>>>(DN0, (size_t)Nn * H);

  {
    dim3 g((Nn + 127) / 128, (H * HID) / 64);
    gemm_f32_wmma<<<g, blk, IN * 64 * sizeof(float), stream>>>(x, W0, H0buf, Nn, H * HID, IN);
  }
  node_scores<<<blocks_for((long long)Nn * H * 32, 256), blk, 0, stream>>>(
      H0buf, asrc0, adst0, SS0, SD0, Nn, H, HID);
  edge_segmax<<<blocks_for(ET * H, 256), blk, 0, stream>>>(ei, E, Nn, SS0, SD0, EM0, H);
  edge_expsum<<<blocks_for(ET * H, 256), blk, 0, stream>>>(ei, E, Nn, SS0, SD0, EM0, W0b, DN0, H);
  edge_norm<<<blocks_for(ET * H, 256), blk, 0, stream>>>(ei, E, Nn, W0b, DN0, H);
  aggregate<<<blocks_for(ET * H * HID, 256), blk, 0, stream>>>(ei, E, Nn, H0buf, W0b, OUT0, H, HID);
  bias_elu_inplace<<<blocks_for((long long)Nn * H * HID, 256), blk, 0, stream>>>(
      OUT0, b0, (size_t)Nn * H * HID, H * HID);

  // ===== layer 1 (H=1, concat=False means identity mean) =====
  zero_f32<<<blocks_for((long long)Nn * OUTF, 256), blk, 0, stream>>>(OUT1, (size_t)Nn * OUTF);
  zero_f32<<<blocks_for((long long)Nn, 256), blk, 0, stream>>>((float*)EM1, (size_t)Nn);
  zero_f32<<<blocks_for((long long)Nn, 256), blk, 0, stream>>>(DN1, (size_t)Nn);

  {
    dim3 g((Nn + 127) / 128, OUTF / 64);
    gemm_f32_wmma<<<g, blk, (H * HID) * 64 * sizeof(float), stream>>>(OUT0, W1, H1buf, Nn, OUTF, H * HID);
  }
  node_scores<<<blocks_for((long long)Nn * 32, 256), blk, 0, stream>>>(
      H1buf, asrc1, adst1, SS1, SD1, Nn, 1, OUTF);
  edge_segmax<<<blocks_for(ET, 256), blk, 0, stream>>>(ei, E, Nn, SS1, SD1, EM1, 1);
  edge_expsum<<<blocks_for(ET, 256), blk, 0, stream>>>(ei, E, Nn, SS1, SD1, EM1, W1b, DN1, 1);
  edge_norm<<<blocks_for(ET, 256), blk, 0, stream>>>(ei, E, Nn, W1b, DN1, 1);
  aggregate<<<blocks_for(ET * OUTF, 256), blk, 0, stream>>>(ei, E, Nn, H1buf, W1b, OUT1, 1, OUTF);

  residual_out<<<blocks_for((long long)Nn * OUTF, 256), blk, 0, stream>>>(
      x, OUT1, b1, (float*)d_out, (size_t)Nn * OUTF, OUTF);
}